// ESN_21010980012343
// MI455X (gfx1250) — compile-verified
//
#include <hip/hip_runtime.h>
#include <math.h>

#define N   4096
#define T   2000
#define SR  0.9f     // spectral radius
#define LK  0.7f     // leaky rate

typedef __attribute__((ext_vector_type(2))) float v2f;
typedef __attribute__((ext_vector_type(8))) float v8f;

#define NBLK          128
#define NTHR          256            // 8 waves of 32 (wave32)
#define ROWS_PER_BLK  32             // 2 row-tiles of 16
#define KSPLIT        4              // 4 waves cooperate on one row-tile's K
#define KCHUNK        (N / KSPLIT)   // 1024

// ---- init: extract w_in0 = W_in[:,0], reset grid-barrier counter ----------
__global__ __launch_bounds__(NTHR)
void esn_init(const float* __restrict__ Win,
              float* __restrict__ win0,
              unsigned* __restrict__ cnt) {
    int i = blockIdx.x * blockDim.x + threadIdx.x;
    if (i < N) win0[i] = Win[(size_t)i * N];     // column 0 of row-major W_in
    if (i == 0)
        __hip_atomic_store(cnt, 0u, __ATOMIC_RELAXED, __HIP_MEMORY_SCOPE_AGENT);
}

// ---- persistent kernel: all T steps, grid barrier between steps -----------
__global__ __launch_bounds__(NTHR)
void esn_persistent(const float* __restrict__ W,
                    const float* __restrict__ u,
                    const float* __restrict__ x0,
                    const float* __restrict__ win0,
                    float* __restrict__ out,
                    unsigned* __restrict__ cnt) {
    __shared__ float xs[N];          // current state vector (16 KB)
    __shared__ float yred[8][16];    // per-wave partial y for reduction

    const int tid  = threadIdx.x;
    const int bid  = blockIdx.x;
    const int lane = tid & 31;
    const int wid  = tid >> 5;       // 0..7
    const int m    = lane & 15;      // row within 16-row tile (A layout)
    const int hi   = lane >> 4;      // half-wave -> K slots {2hi, 2hi+1}
    const int rt   = wid >> 2;       // row tile within block: 0..1
    const int q    = wid & 3;        // K quarter: 0..3
    const int row0 = bid * ROWS_PER_BLK + rt * 16;
    const int kbase = q * KCHUNK;

    // per-lane base pointers, pre-offset by this wave's K quarter
    const float* Ap = W + (size_t)(row0 + m) * N + (size_t)(kbase + 2 * hi);
    const float* xp = &xs[kbase + 2 * hi];

    // LDS byte offset of xs (low 32 bits of the generic address == LDS offset)
    const unsigned xs_lds = (unsigned)(uintptr_t)(&xs[0]);

    for (int t = 0; t < T; ++t) {
        // ---- stage x_{t-1} into LDS via async copy (row t-1; x0 for t==0)
        {
            const float* xsrc = (t == 0) ? x0 : (out + (size_t)(t - 1) * N);
            #pragma unroll
            for (int c = 0; c < 4; ++c) {
                int i = (tid + c * NTHR) * 4;            // float index, 16B chunk
                unsigned dlds = xs_lds + (unsigned)(i * 4);
                const float* gaddr = xsrc + i;
                asm volatile("global_load_async_to_lds_b128 %0, %1, off"
                             :: "v"(dlds), "v"(gaddr) : "memory");
            }
            asm volatile("s_wait_asynccnt 0x0" ::: "memory");
        }
        __syncthreads();

        // ---- WMMA matvec: 16 rows x KCHUNK, 4 independent accumulators ---
        v8f a0 = {}, a1 = {}, a2 = {}, a3 = {};
        #pragma unroll 2
        for (int kk = 0; kk < KCHUNK; kk += 16) {        // constant trip count
            v2f A0 = *(const v2f*)(Ap + kk);
            v2f A1 = *(const v2f*)(Ap + kk + 4);
            v2f A2 = *(const v2f*)(Ap + kk + 8);
            v2f A3 = *(const v2f*)(Ap + kk + 12);
            // B: every column = x chunk; per-lane K slots {2*hi, 2*hi+1}
            v2f B0 = *(const v2f*)(xp + kk);
            v2f B1 = *(const v2f*)(xp + kk + 4);
            v2f B2 = *(const v2f*)(xp + kk + 8);
            v2f B3 = *(const v2f*)(xp + kk + 12);
            a0 = __builtin_amdgcn_wmma_f32_16x16x4_f32(false, A0, false, B0,
                                                       (short)0, a0, false, false);
            a1 = __builtin_amdgcn_wmma_f32_16x16x4_f32(false, A1, false, B1,
                                                       (short)0, a1, false, false);
            a2 = __builtin_amdgcn_wmma_f32_16x16x4_f32(false, A2, false, B2,
                                                       (short)0, a2, false, false);
            a3 = __builtin_amdgcn_wmma_f32_16x16x4_f32(false, A3, false, B3,
                                                       (short)0, a3, false, false);
        }
        v8f acc = (a0 + a1) + (a2 + a3);

        // D layout: VGPR r, lanes 0-15 -> row M=r, lanes 16-31 -> M=8+r.
        // Columns are identical, so lane 0 / lane 16 hold the 16 partial y's.
        if (m == 0) {
            #pragma unroll
            for (int r = 0; r < 8; ++r) yred[wid][8 * hi + r] = acc[r];
        }
        __syncthreads();

        // ---- reduce K quarters + elementwise update + store state row ----
        if (tid < ROWS_PER_BLK) {
            int r   = tid;
            int rtt = r >> 4;
            int idx = r & 15;
            float y = yred[rtt * 4 + 0][idx] + yred[rtt * 4 + 1][idx]
                    + yred[rtt * 4 + 2][idx] + yred[rtt * 4 + 3][idx];
            int g = bid * ROWS_PER_BLK + r;
            float pre = SR * y + win0[g] * u[t];
            float xn  = (1.0f - LK) * xs[g] + LK * tanhf(pre);
            out[(size_t)t * N + g] = xn;
        }

        // ---- device-wide barrier (monotonic counter, no reset/ABA) -------
        __threadfence();                       // release: state row visible
        __syncthreads();
        if (tid == 0) {
            __hip_atomic_fetch_add(cnt, 1u, __ATOMIC_RELAXED,
                                   __HIP_MEMORY_SCOPE_AGENT);
            unsigned target = (unsigned)(t + 1) * (unsigned)NBLK;
            while (__hip_atomic_load(cnt, __ATOMIC_RELAXED,
                                     __HIP_MEMORY_SCOPE_AGENT) < target)
                __builtin_amdgcn_s_sleep(1);
        }
        __syncthreads();
        __threadfence();                       // acquire: invalidate L0
    }
}

// ---------------------------------------------------------------------------
extern "C" void kernel_launch(void* const* d_in, const int* in_sizes, int n_in,
                              void* d_out, int out_size, void* d_ws, size_t ws_size,
                              hipStream_t stream) {
    const float* x0  = (const float*)d_in[0];   // (N,)
    const float* u   = (const float*)d_in[1];   // (T,)
    const float* W   = (const float*)d_in[2];   // (N,N)
    const float* Win = (const float*)d_in[3];   // (N,N)
    float* out = (float*)d_out;                 // (T,N)

    unsigned* cnt = (unsigned*)d_ws;                    // barrier counter
    float* win0   = (float*)((char*)d_ws + 256);        // w_in0 (N floats)

    esn_init<<<(N + NTHR - 1) / NTHR, NTHR, 0, stream>>>(Win, win0, cnt);
    esn_persistent<<<NBLK, NTHR, 0, stream>>>(W, u, x0, win0, out, cnt);
}